// BitInput_17849884082284
// MI455X (gfx1250) — compile-verified
//
#include <hip/hip_runtime.h>
#include <stdint.h>

// out[e][j] = (perm_e(j) < round(x[e]*256)) ? 1.0f : 0.0f,  e<262144, j<256.
// Output = 256 MB f32 -> HBM-store bound (~11 us @ 23.3 TB/s). perm_e is a keyed
// 8-bit bijection (affine mod 256 o xorshift, twice) -> exactly round(p*256) ones
// at pseudo-random positions (reference semantics: exact count, random placement).
//
// MI455X shaping:
//  * bijection evaluated in 2-way packed u16 (VOP3P v_pk_mad_u16 / v_pk_lshrrev_b16)
//    -> ~4 VALU/output instead of ~10 scalar; cmp+cndmask materialization (2/output)
//    is the VCC floor.
//  * 32 outputs/thread (8 x float4) so the per-element key hash (8 threads/element)
//    and scaffolding amortize; ~270 inst/wave for 4 KB ~= issue-balanced vs BW floor.
//  * CDNA5 data path: ds_store_b128 staging -> s_wait_dscnt ->
//    global_store_async_from_lds_b128 (ASYNCcnt) -> s_wait_asynccnt.

#define THREADS_PER_BLOCK 256
#define CHUNKS_PER_THREAD 8            // 8 x float4 = 32 outputs per thread
#define THREADS_PER_ELEM  8            // 8 threads x 32 outputs = 256 bits/element
#define ELEMS_PER_BLOCK   32           // 256 threads / 8

typedef int v4i __attribute__((vector_size(16)));
typedef unsigned short v2u16 __attribute__((ext_vector_type(2)));

__global__ __launch_bounds__(THREADS_PER_BLOCK)
void bitstream256_kernel(const float* __restrict__ x,
                         float* __restrict__ out,
                         int n_elems) {
    __shared__ float4 smem[THREADS_PER_BLOCK * CHUNKS_PER_THREAD];   // 32 KB staging

    const int t    = threadIdx.x;
    const int sub  = t & (THREADS_PER_ELEM - 1);   // 0..7 within element group
    const int egrp = t >> 3;                       // 0..31 element slot in block
    const int e    = blockIdx.x * ELEMS_PER_BLOCK + egrp;
    if (e >= n_elems) return;

    // Prefetch input ahead (global_prefetch_b8); input is L2-resident anyway.
    __builtin_prefetch(x + e + 8192, 0, 1);

    const float p = x[e];
    int n = __float2int_rn(p * 256.0f);            // n_ones, clamped to [0,256]
    n = n < 0 ? 0 : (n > 256 ? 256 : n);
    const unsigned short un = (unsigned short)n;

    // Per-element key (murmur3-style finalizer); amortized over 32 outputs/thread.
    uint32_t k = (uint32_t)e * 0x9E3779B1u;
    k ^= k >> 15; k *= 0x85EBCA77u;
    k ^= k >> 13; k *= 0xC2B2AE3Du;
    k ^= k >> 16;
    const unsigned short a1 = (unsigned short)((k & 0xFFu) | 1u);          // odd
    const unsigned short b1 = (unsigned short)((k >> 8)  & 0xFFu);
    const unsigned short a2 = (unsigned short)(((k >> 16) & 0xFFu) | 1u); // odd
    const unsigned short b2 = (unsigned short)((k >> 24) & 0xFFu);

    const v2u16 va1   = {a1, a1}, vb1 = {b1, b1};
    const v2u16 va2   = {a2, a2}, vb2 = {b2, b2};
    const v2u16 vmask = {0xFF, 0xFF};
    const v2u16 vsh4  = {4, 4};
    const v2u16 vsh3  = {3, 3};

    // Stage 8 float4 chunks: chunk c holds bits j = sub*4 + c*32 + {0..3}
#pragma unroll
    for (int c = 0; c < CHUNKS_PER_THREAD; ++c) {
        const unsigned short jb = (unsigned short)(sub * 4 + c * 32);
        float4 v;
        float* vf = reinterpret_cast<float*>(&v);
#pragma unroll
        for (int h = 0; h < 2; ++h) {              // two packed pairs -> 4 outputs
            v2u16 r = {(unsigned short)(jb + 2 * h),
                       (unsigned short)(jb + 2 * h + 1)};
            r = (v2u16)(r * va1 + vb1) & vmask;    // packed affine mod 256 (bijective)
            r ^= r >> vsh4;                        // packed xorshift (bijective)
            r = (v2u16)(r * va2 + vb2) & vmask;
            r ^= r >> vsh3;
            vf[2 * h + 0] = (r.x < un) ? 1.0f : 0.0f;
            vf[2 * h + 1] = (r.y < un) ? 1.0f : 0.0f;
        }
        smem[t * CHUNKS_PER_THREAD + c] = v;       // ds_store_b128
    }

    // One LDS-visibility wait for all staged chunks (per-wave DScnt).
#if __has_builtin(__builtin_amdgcn_s_wait_dscnt)
    __builtin_amdgcn_s_wait_dscnt(0);
#else
    asm volatile("s_wait_dscnt 0x0" ::: "memory");
#endif

    // Drain LDS -> HBM with the async store engine. Per instruction, each group of
    // 8 lanes writes a contiguous 128B segment of one element (4 elements/wave).
    float* gbase = out + (size_t)e * 256 + sub * 4;
#pragma unroll
    for (int c = 0; c < CHUNKS_PER_THREAD; ++c) {
        float* gdst = gbase + c * 32;
#if __has_builtin(__builtin_amdgcn_global_store_async_from_lds_b128)
        typedef __attribute__((address_space(1))) v4i gv4i;
        typedef __attribute__((address_space(3))) v4i lv4i;
        gv4i* gp = (gv4i*)(uintptr_t)gdst;
        // AS(3) pointer value == LDS offset == low 32 bits of the generic pointer
        lv4i* lp = (lv4i*)(uint32_t)(uintptr_t)(&smem[t * CHUNKS_PER_THREAD + c]);
        __builtin_amdgcn_global_store_async_from_lds_b128(gp, lp, /*offset=*/0, /*cpol=*/0);
#else
        uint64_t ga = (uint64_t)(uintptr_t)gdst;
        uint32_t la = (uint32_t)(uintptr_t)(&smem[t * CHUNKS_PER_THREAD + c]);
        asm volatile("global_store_async_from_lds_b128 %0, %1, off"
                     :: "v"(ga), "v"(la) : "memory");
#endif
    }

    // Drain async stores before wave exit (S_ENDPGM also implies wait-idle).
#if __has_builtin(__builtin_amdgcn_s_wait_asynccnt)
    __builtin_amdgcn_s_wait_asynccnt(0);
#else
    asm volatile("s_wait_asynccnt 0x0" ::: "memory");
#endif
}

extern "C" void kernel_launch(void* const* d_in, const int* in_sizes, int n_in,
                              void* d_out, int out_size, void* d_ws, size_t ws_size,
                              hipStream_t stream) {
    (void)d_ws; (void)ws_size; (void)n_in;
    const float* x = (const float*)d_in[0];
    float* out = (float*)d_out;

    // out_size = n_elems * 256 (bit_size); kernel is specialized for N=256.
    const int n_elems = out_size >> 8;
    const int blocks  = (n_elems + ELEMS_PER_BLOCK - 1) / ELEMS_PER_BLOCK;

    bitstream256_kernel<<<blocks, THREADS_PER_BLOCK, 0, stream>>>(x, out, n_elems);
}